// HyperKA_59030030516651
// MI455X (gfx1250) — compile-verified
//
#include <hip/hip_runtime.h>
#include <math.h>

#define EPSC 1e-5f
#define MAXN (1.0f - 1e-5f)
#define DDIM 128

typedef __attribute__((ext_vector_type(2))) float v2f;
typedef __attribute__((ext_vector_type(8))) float v8f;

// ---------------- wave32 helpers ----------------
__device__ __forceinline__ float wave_sum(float v) {
  v += __shfl_xor(v, 16, 32);
  v += __shfl_xor(v, 8, 32);
  v += __shfl_xor(v, 4, 32);
  v += __shfl_xor(v, 2, 32);
  v += __shfl_xor(v, 1, 32);
  return v;
}
__device__ __forceinline__ float dot4(const float4& a, const float4& b) {
  return a.x * b.x + a.y * b.y + a.z * b.z + a.w * b.w;
}
__device__ __forceinline__ float4 scale4(const float4& a, float s) {
  return make_float4(a.x * s, a.y * s, a.z * s, a.w * s);
}
__device__ __forceinline__ float4 add4(const float4& a, const float4& b) {
  return make_float4(a.x + b.x, a.y + b.y, a.z + b.z, a.w + b.w);
}
// row norm across one wave (each lane holds 4 of 128 elems)
__device__ __forceinline__ float rownorm(const float4& x) {
  return sqrtf(fmaxf(wave_sum(dot4(x, x)), 1e-12f));
}
__device__ __forceinline__ float4 log0(const float4& x) {
  float n = rownorm(x);
  float nc = fminf(fmaxf(n, EPSC), MAXN);
  float s = atanhf(nc) / nc;
  return scale4(x, s);
}
__device__ __forceinline__ float4 exp0(const float4& v) {
  float n = fmaxf(rownorm(v), EPSC);
  float s = tanhf(n) / n;
  return scale4(v, s);
}
__device__ __forceinline__ float4 hproj(const float4& x) {
  float n = rownorm(x);
  return (n >= MAXN) ? scale4(x, MAXN / n) : x;
}
__device__ __forceinline__ float4 mobius(const float4& x, const float4& y) {
  float x2 = wave_sum(dot4(x, x));
  float y2 = wave_sum(dot4(y, y));
  float xy = wave_sum(dot4(x, y));
  float ca = 1.0f + 2.0f * xy + y2;
  float cb = 1.0f - x2;
  float4 num = add4(scale4(x, ca), scale4(y, cb));
  float den = 1.0f + 2.0f * xy + x2 * y2;
  float sg = (den > 0.0f) ? 1.0f : ((den < 0.0f) ? -1.0f : 0.0f);
  float inv = sg / fmaxf(fabsf(den), EPSC);
  return scale4(num, inv);
}
// order-preserving float <-> uint for atomic segment-max
__device__ __forceinline__ unsigned fenc(float f) {
  unsigned u = __float_as_uint(f);
  return (u & 0x80000000u) ? ~u : (u | 0x80000000u);
}
__device__ __forceinline__ float fdec(unsigned e) {
  return (e & 0x80000000u) ? __uint_as_float(e ^ 0x80000000u) : __uint_as_float(~e);
}

// ---------------- fill kernels ----------------
__global__ void fill_f32(float* __restrict__ p, float v, size_t n) {
  size_t i = (size_t)blockIdx.x * blockDim.x + threadIdx.x;
  if (i < n) p[i] = v;
}
__global__ void fill_u32(unsigned* __restrict__ p, unsigned v, size_t n) {
  size_t i = (size_t)blockIdx.x * blockDim.x + threadIdx.x;
  if (i < n) p[i] = v;
}

// ---------------- log_map_zero over rows ----------------
__global__ void logmap_rows(const float* __restrict__ in, float* __restrict__ out, int nrows) {
  int row = blockIdx.x * (blockDim.x >> 5) + (threadIdx.x >> 5);
  int lane = threadIdx.x & 31;
  if (row >= nrows) return;
  float4 x = ((const float4*)(in + (size_t)row * DDIM))[lane];
  float4 o = log0(x);
  ((float4*)(out + (size_t)row * DDIM))[lane] = o;
}

// ---------------- WMMA fp32 GEMM: C[nrows,128] = A[nrows,128] * B[128,128] ----------------
// One wave per 16-row strip: 8 accumulators cover all 128 output columns.
// A fragment loaded once per K-step (global_load_b64), 8 v_wmma_f32_16x16x4_f32 per step.
__global__ void __launch_bounds__(32) gemm128_wmma(const float* __restrict__ A,
                                                   const float* __restrict__ B,
                                                   float* __restrict__ C, int nrows) {
  const int lane = threadIdx.x;
  const int half = lane >> 4;   // 0: K pair {0,1}, 1: K pair {2,3}
  const int lm = lane & 15;
  const int r0 = blockIdx.x * 16;

  v8f acc0 = {}, acc1 = {}, acc2 = {}, acc3 = {};
  v8f acc4 = {}, acc5 = {}, acc6 = {}, acc7 = {};

  // clamp OOB A rows to a valid row: they only feed OOB C rows (store-masked)
  int arow = r0 + lm;
  if (arow >= nrows) arow = nrows - 1;
  const float* Ar = A + (size_t)arow * DDIM;
  const float* B0 = B + lm;

  for (int k = 0; k < DDIM; k += 4) {
    const int ka = k + 2 * half;
    // A 16x4 fragment: lanes 0-15 hold K={ka,ka+1} of row lm; lanes 16-31 K={ka,ka+1} (pair 2,3)
    const v2f a = *(const v2f*)(Ar + ka);
    const float* Brow0 = B0 + (size_t)ka * DDIM;
    const float* Brow1 = Brow0 + DDIM;
#pragma unroll
    for (int c = 0; c < 8; ++c) {
      v2f b;
      b.x = Brow0[c * 16];
      b.y = Brow1[c * 16];
      switch (c) {
        case 0: acc0 = __builtin_amdgcn_wmma_f32_16x16x4_f32(false, a, false, b, (short)0, acc0, false, false); break;
        case 1: acc1 = __builtin_amdgcn_wmma_f32_16x16x4_f32(false, a, false, b, (short)0, acc1, false, false); break;
        case 2: acc2 = __builtin_amdgcn_wmma_f32_16x16x4_f32(false, a, false, b, (short)0, acc2, false, false); break;
        case 3: acc3 = __builtin_amdgcn_wmma_f32_16x16x4_f32(false, a, false, b, (short)0, acc3, false, false); break;
        case 4: acc4 = __builtin_amdgcn_wmma_f32_16x16x4_f32(false, a, false, b, (short)0, acc4, false, false); break;
        case 5: acc5 = __builtin_amdgcn_wmma_f32_16x16x4_f32(false, a, false, b, (short)0, acc5, false, false); break;
        case 6: acc6 = __builtin_amdgcn_wmma_f32_16x16x4_f32(false, a, false, b, (short)0, acc6, false, false); break;
        case 7: acc7 = __builtin_amdgcn_wmma_f32_16x16x4_f32(false, a, false, b, (short)0, acc7, false, false); break;
      }
    }
  }

  // C/D 16x16 layout: VGPR v -> M = v + 8*half, N = lane%16 (per column tile)
  const int rbase = r0 + half * 8;
#pragma unroll
  for (int v = 0; v < 8; ++v) {
    const int cr = rbase + v;
    if (cr < nrows) {
      float* Crow = C + (size_t)cr * DDIM + lm;
      Crow[0 * 16] = acc0[v];
      Crow[1 * 16] = acc1[v];
      Crow[2 * 16] = acc2[v];
      Crow[3 * 16] = acc3[v];
      Crow[4 * 16] = acc4[v];
      Crow[5 * 16] = acc5[v];
      Crow[6 * 16] = acc6[v];
      Crow[7 * 16] = acc7[v];
    }
  }
}

// ---------------- edge kernels ----------------
__global__ void edge_scores(const float* __restrict__ em, const int* __restrict__ src,
                            const int* __restrict__ dst, float* __restrict__ s, int E) {
  int e = blockIdx.x * (blockDim.x >> 5) + (threadIdx.x >> 5);
  int lane = threadIdx.x & 31;
  if (e >= E) return;
  float4 a = ((const float4*)(em + (size_t)src[e] * DDIM))[lane];
  float4 b = ((const float4*)(em + (size_t)dst[e] * DDIM))[lane];
  float d = wave_sum(dot4(a, b));
  if (lane == 0) s[e] = d;
}

__global__ void edge_segmax(const float* __restrict__ s, const int* __restrict__ src,
                            unsigned* __restrict__ rmax, int E) {
  int e = blockIdx.x * blockDim.x + threadIdx.x;
  if (e >= E) return;
  atomicMax(&rmax[src[e]], fenc(s[e]));
}

__global__ void edge_expsum(const float* __restrict__ s, const int* __restrict__ src,
                            const unsigned* __restrict__ rmax, float* __restrict__ exb,
                            float* __restrict__ rsum, int E) {
  int e = blockIdx.x * blockDim.x + threadIdx.x;
  if (e >= E) return;
  int si = src[e];
  float m = fdec(rmax[si]);
  float ex = expf(s[e] - m);
  exb[e] = ex;
  unsafeAtomicAdd(&rsum[si], ex);
}

__global__ void edge_aggregate(const float* __restrict__ em, const int* __restrict__ src,
                               const int* __restrict__ dst, const float* __restrict__ exb,
                               const float* __restrict__ rsum, float* __restrict__ ee_agg,
                               int E) {
  int e = blockIdx.x * (blockDim.x >> 5) + (threadIdx.x >> 5);
  int lane = threadIdx.x & 31;
  if (e >= E) return;
  int si = src[e], di = dst[e];
  float alpha = exb[e] / fmaxf(rsum[si], EPSC);
  float4 v = ((const float4*)(em + (size_t)di * DDIM))[lane];
  float* p = ee_agg + (size_t)si * DDIM + lane * 4;
  unsafeAtomicAdd(p + 0, alpha * v.x);
  unsafeAtomicAdd(p + 1, alpha * v.y);
  unsafeAtomicAdd(p + 2, alpha * v.z);
  unsafeAtomicAdd(p + 3, alpha * v.w);
}

__global__ void edge_relagg(const float* __restrict__ rm, const int* __restrict__ rel_ent,
                            const int* __restrict__ rel_rel, float* __restrict__ er_agg,
                            float* __restrict__ ent_cnt, float* __restrict__ rel_cnt, int E) {
  int e = blockIdx.x * (blockDim.x >> 5) + (threadIdx.x >> 5);
  int lane = threadIdx.x & 31;
  if (e >= E) return;
  int ei = rel_ent[e], ri = rel_rel[e];
  float4 v = ((const float4*)(rm + (size_t)ri * DDIM))[lane];
  float* p = er_agg + (size_t)ei * DDIM + lane * 4;
  unsafeAtomicAdd(p + 0, v.x);
  unsafeAtomicAdd(p + 1, v.y);
  unsafeAtomicAdd(p + 2, v.z);
  unsafeAtomicAdd(p + 3, v.w);
  if (lane == 0) {
    unsafeAtomicAdd(&ent_cnt[ei], 1.0f);
    unsafeAtomicAdd(&rel_cnt[ri], 1.0f);
  }
}

// ---------------- combine kernels (fused hyperbolic chain, one wave per row) ----------------
__global__ void combine_entity(const float* __restrict__ ee_agg, const float* __restrict__ er_agg,
                               const float* __restrict__ ent_cnt, const float* __restrict__ bias,
                               float* __restrict__ out, int nrows, int use_act) {
  int row = blockIdx.x * (blockDim.x >> 5) + (threadIdx.x >> 5);
  int lane = threadIdx.x & 31;
  if (row >= nrows) return;
  float4 ee = ((const float4*)(ee_agg + (size_t)row * DDIM))[lane];
  float4 er = ((const float4*)(er_agg + (size_t)row * DDIM))[lane];
  float cnt = fmaxf(ent_cnt[row], 1.0f);
  er = scale4(er, 1.0f / cnt);
  float4 eeh = hproj(exp0(ee));
  float4 erh = hproj(exp0(er));
  float4 oe = hproj(mobius(eeh, scale4(erh, 0.1f)));
  float4 bv = ((const float4*)bias)[lane];
  float4 bh = hproj(exp0(bv));
  oe = hproj(mobius(oe, bh));
  if (use_act) {
    float4 t = log0(oe);
    t = make_float4(tanhf(t.x), tanhf(t.y), tanhf(t.z), tanhf(t.w));
    oe = hproj(exp0(t));
  }
  ((float4*)(out + (size_t)row * DDIM))[lane] = oe;
}

__global__ void combine_rel(const float* __restrict__ rt, const float* __restrict__ rel_cnt,
                            float* __restrict__ out, int nrows, int use_act) {
  int row = blockIdx.x * (blockDim.x >> 5) + (threadIdx.x >> 5);
  int lane = threadIdx.x & 31;
  if (row >= nrows) return;
  float4 v = ((const float4*)(rt + (size_t)row * DDIM))[lane];
  float cnt = rel_cnt[row];
  float4 mean = (cnt > 0.0f) ? v : make_float4(0.f, 0.f, 0.f, 0.f);
  float4 orr = hproj(exp0(mean));
  if (use_act) {
    float4 t = log0(orr);
    t = make_float4(tanhf(t.x), tanhf(t.y), tanhf(t.z), tanhf(t.w));
    orr = hproj(exp0(t));
  }
  ((float4*)(out + (size_t)row * DDIM))[lane] = orr;
}

// ---------------- host launcher ----------------
extern "C" void kernel_launch(void* const* d_in, const int* in_sizes, int n_in,
                              void* d_out, int out_size, void* d_ws, size_t ws_size,
                              hipStream_t stream) {
  (void)n_in; (void)out_size; (void)ws_size;
  const float* ents  = (const float*)d_in[0];
  const float* rels  = (const float*)d_in[1];
  const float* W_ent = (const float*)d_in[2];
  const float* W_rel = (const float*)d_in[3];
  const float* bias  = (const float*)d_in[4];
  const int* ent_src = (const int*)d_in[5];
  const int* ent_dst = (const int*)d_in[6];
  const int* rel_ent = (const int*)d_in[7];
  const int* rel_rel = (const int*)d_in[8];

  const int N = in_sizes[0] / DDIM;
  const int R = in_sizes[1] / DDIM;
  const int L = in_sizes[2] / (DDIM * DDIM);
  const int E = in_sizes[5];

  // workspace carve (256B aligned)
  char* base = (char*)d_ws;
  size_t off = 0;
  auto alloc = [&](size_t bytes) -> void* {
    void* p = base + off;
    off += (bytes + 255) & ~(size_t)255;
    return p;
  };
  const size_t ND = (size_t)N * DDIM, RD = (size_t)R * DDIM;
  float* et     = (float*)alloc(ND * 4);
  float* em     = (float*)alloc(ND * 4);
  float* ee_agg = (float*)alloc(ND * 4);
  float* er_agg = (float*)alloc(ND * 4);
  float* e_mid  = (float*)alloc(ND * 4);
  float* rt     = (float*)alloc(RD * 4);
  float* rm     = (float*)alloc(RD * 4);
  float* r_mid  = (float*)alloc(RD * 4);
  float* sbuf   = (float*)alloc((size_t)E * 4);
  float* exb    = (float*)alloc((size_t)E * 4);
  unsigned* rmaxu = (unsigned*)alloc((size_t)N * 4);
  float* rsum   = (float*)alloc((size_t)N * 4);
  float* entcnt = (float*)alloc((size_t)N * 4);
  float* relcnt = (float*)alloc((size_t)R * 4);

  float* out_e = (float*)d_out;
  float* out_r = out_e + ND;

  const float* ecur = ents;
  const float* rcur = rels;

  const int WPB = 8;  // waves per 256-thread block

  for (int l = 0; l < L; ++l) {
    const int use_act = (l < L - 1) ? 1 : 0;
    const float* We = W_ent + (size_t)l * DDIM * DDIM;
    const float* Wr = W_rel + (size_t)l * DDIM * DDIM;
    const float* bl = bias + (size_t)l * DDIM;
    float* eo = (l == L - 1) ? out_e : e_mid;
    float* ro = (l == L - 1) ? out_r : r_mid;

    // tangent-space projections
    logmap_rows<<<(N + WPB - 1) / WPB, 256, 0, stream>>>(ecur, et, N);
    logmap_rows<<<(R + WPB - 1) / WPB, 256, 0, stream>>>(rcur, rt, R);

    // WMMA GEMMs: one wave per 16-row strip
    gemm128_wmma<<<(N + 15) / 16, 32, 0, stream>>>(et, We, em, N);
    gemm128_wmma<<<(R + 15) / 16, 32, 0, stream>>>(rt, Wr, rm, R);

    // zero accumulators (ws is not re-poisoned between replays)
    fill_f32<<<(unsigned)((ND + 255) / 256), 256, 0, stream>>>(ee_agg, 0.f, ND);
    fill_f32<<<(unsigned)((ND + 255) / 256), 256, 0, stream>>>(er_agg, 0.f, ND);
    fill_u32<<<(N + 255) / 256, 256, 0, stream>>>(rmaxu, 0u, (size_t)N);
    fill_f32<<<(N + 255) / 256, 256, 0, stream>>>(rsum, 0.f, (size_t)N);
    fill_f32<<<(N + 255) / 256, 256, 0, stream>>>(entcnt, 0.f, (size_t)N);
    fill_f32<<<(R + 255) / 256, 256, 0, stream>>>(relcnt, 0.f, (size_t)R);

    // sparse attention
    edge_scores<<<(E + WPB - 1) / WPB, 256, 0, stream>>>(em, ent_src, ent_dst, sbuf, E);
    edge_segmax<<<(E + 255) / 256, 256, 0, stream>>>(sbuf, ent_src, rmaxu, E);
    edge_expsum<<<(E + 255) / 256, 256, 0, stream>>>(sbuf, ent_src, rmaxu, exb, rsum, E);
    edge_aggregate<<<(E + WPB - 1) / WPB, 256, 0, stream>>>(em, ent_src, ent_dst, exb, rsum,
                                                            ee_agg, E);
    edge_relagg<<<(E + WPB - 1) / WPB, 256, 0, stream>>>(rm, rel_ent, rel_rel, er_agg,
                                                         entcnt, relcnt, E);

    // fused hyperbolic combine
    combine_entity<<<(N + WPB - 1) / WPB, 256, 0, stream>>>(ee_agg, er_agg, entcnt, bl, eo, N,
                                                            use_act);
    combine_rel<<<(R + WPB - 1) / WPB, 256, 0, stream>>>(rt, relcnt, ro, R, use_act);

    ecur = eo;
    rcur = ro;
  }
}